// ResHybridBlock_75771813036526
// MI455X (gfx1250) — compile-verified
//
#include <hip/hip_runtime.h>
#include <math.h>

// Problem constants (match reference)
#define NN  20000
#define EE  320000
#define ESL 340000      // E + N self loops
#define CC  128
#define HH  4
#define HC  512         // H*C
#define BN_EPS 1e-5f

typedef __attribute__((ext_vector_type(2))) float v2f;
typedef __attribute__((ext_vector_type(8))) float v8f;

// ---------------- workspace layout (in floats) ----------------
#define OFF_H     ((size_t)0)                       // [N, HC]  GAT pre-agg features
#define OFF_GAT   (OFF_H   + (size_t)NN*HC)         // [N, HC]  GAT aggregated output
#define OFF_AS    (OFF_GAT + (size_t)NN*HC)         // [N, H]   alpha_src
#define OFF_AD    (OFF_AS  + (size_t)NN*HH)         // [N, H]   alpha_dst
#define OFF_MU    (OFF_AD  + (size_t)NN*HH)         // [N, H]   segment max (sortable uint)
#define OFF_DEN   (OFF_MU  + (size_t)NN*HH)         // [N, H]   softmax denom
#define OFF_DEG   (OFF_DEN + (size_t)NN*HH)         // [N]      in-degree (incl self loop)
#define OFF_SUM1  (OFF_DEG + (size_t)NN)            // [HC]
#define OFF_SQ1   (OFF_SUM1 + (size_t)HC)           // [HC]
#define OFF_SC1   (OFF_SQ1  + (size_t)HC)           // [HC] bn1 scale
#define OFF_SH1   (OFF_SC1  + (size_t)HC)           // [HC] bn1 shift (bias folded)
#define OFF_XW    (OFF_SH1  + (size_t)HC)           // [N, C]  GCN xw
#define OFF_GCN   (OFF_XW   + (size_t)NN*CC)        // [N, C]  GCN aggregated
#define OFF_SUM2  (OFF_GCN  + (size_t)NN*CC)        // [C]
#define OFF_SQ2   (OFF_SUM2 + (size_t)CC)           // [C]
#define OFF_SC2   (OFF_SQ2  + (size_t)CC)           // [C]
#define OFF_SH2   (OFF_SC2  + (size_t)CC)           // [C]
#define WS_FLOATS (OFF_SH2  + (size_t)CC)

// ---------------- helpers ----------------
__device__ __forceinline__ unsigned enc_f(float f) {
  unsigned u = __float_as_uint(f);
  return (u & 0x80000000u) ? ~u : (u | 0x80000000u);
}
__device__ __forceinline__ float dec_f(unsigned u) {
  return __uint_as_float((u & 0x80000000u) ? (u & 0x7FFFFFFFu) : ~u);
}
__device__ __forceinline__ void edge_sd(const int* __restrict__ ei, int e, int& s, int& d) {
  if (e < EE) { s = ei[e]; d = ei[EE + e]; }
  else        { s = e - EE; d = e - EE; }        // appended self loops
}
__device__ __forceinline__ float lrelu(float v) { return v > 0.f ? v : 0.2f * v; }

// ---------------- zero ----------------
__global__ void zero_kernel(float* __restrict__ p, size_t n) {
  size_t i = (size_t)blockIdx.x * blockDim.x + threadIdx.x;
  size_t st = (size_t)gridDim.x * blockDim.x;
  for (; i < n; i += st) p[i] = 0.0f;
}

// ---------------- GEMM1: H = X[20000,128] @ W[128,512] (fp32 WMMA) ----------------
// grid.x = 1250 (M tiles of 16), block = 256 (8 waves); wave w covers cols [64w, 64w+64)
__global__ __launch_bounds__(256) void gemm1_kernel(const float* __restrict__ X,
                                                    const float* __restrict__ W,
                                                    float* __restrict__ Hout) {
  const int wave = threadIdx.x >> 5;
  const int lane = threadIdx.x & 31;
  const int lmod = lane & 15;
  const int lhi  = lane >> 4;
  const int m0 = blockIdx.x * 16;
  const int j0 = wave * 64;

  v8f c0 = {}, c1 = {}, c2 = {}, c3 = {};
  const float* arow = X + (size_t)(m0 + lmod) * CC;
  const int col = j0 + lmod;

  for (int kk = 0; kk < CC; kk += 4) {
    const int kA = kk + 2 * lhi;
    v2f a; a[0] = arow[kA]; a[1] = arow[kA + 1];
    const float* b0p = W + (size_t)kA * HC;
    const float* b1p = W + (size_t)(kA + 1) * HC;
    v2f b;
    b[0] = b0p[col];      b[1] = b1p[col];
    c0 = __builtin_amdgcn_wmma_f32_16x16x4_f32(false, a, false, b, (short)0, c0, false, false);
    b[0] = b0p[col + 16]; b[1] = b1p[col + 16];
    c1 = __builtin_amdgcn_wmma_f32_16x16x4_f32(false, a, false, b, (short)0, c1, false, false);
    b[0] = b0p[col + 32]; b[1] = b1p[col + 32];
    c2 = __builtin_amdgcn_wmma_f32_16x16x4_f32(false, a, false, b, (short)0, c2, false, false);
    b[0] = b0p[col + 48]; b[1] = b1p[col + 48];
    c3 = __builtin_amdgcn_wmma_f32_16x16x4_f32(false, a, false, b, (short)0, c3, false, false);
  }
  const int row0 = m0 + 8 * lhi;
#pragma unroll
  for (int r = 0; r < 8; ++r) {
    float* o = Hout + (size_t)(row0 + r) * HC + col;
    o[0]  = c0[r];
    o[16] = c1[r];
    o[32] = c2[r];
    o[48] = c3[r];
  }
}

// ---------------- attention coefficients per (node, head) ----------------
__global__ void alphas_kernel(const float* __restrict__ Hm,
                              const float* __restrict__ aw_s,
                              const float* __restrict__ aw_d,
                              float* __restrict__ As, float* __restrict__ Ad) {
  int t = blockIdx.x * blockDim.x + threadIdx.x;
  if (t >= NN * HH) return;
  int n = t >> 2, h = t & 3;
  const float* hp = Hm + (size_t)n * HC + h * CC;
  const float* ws = aw_s + h * CC;
  const float* wd = aw_d + h * CC;
  float ss = 0.f, sd = 0.f;
  for (int c = 0; c < CC; ++c) { float v = hp[c]; ss += v * ws[c]; sd += v * wd[c]; }
  As[t] = ss; Ad[t] = sd;
}

// ---------------- edge pass 1: segment max + degree ----------------
__global__ void edge_max_kernel(const int* __restrict__ ei,
                                const float* __restrict__ As, const float* __restrict__ Ad,
                                unsigned* __restrict__ Mu, float* __restrict__ deg) {
  int t = blockIdx.x * blockDim.x + threadIdx.x;
  if (t >= ESL * HH) return;
  int e = t >> 2, h = t & 3;
  int s, d; edge_sd(ei, e, s, d);
  float v = lrelu(As[s * HH + h] + Ad[d * HH + h]);
  atomicMax(&Mu[d * HH + h], enc_f(v));
  if (h == 0) atomicAdd(&deg[d], 1.0f);
}

// ---------------- edge pass 2: softmax denominator ----------------
__global__ void edge_denom_kernel(const int* __restrict__ ei,
                                  const float* __restrict__ As, const float* __restrict__ Ad,
                                  const unsigned* __restrict__ Mu, float* __restrict__ Den) {
  int t = blockIdx.x * blockDim.x + threadIdx.x;
  if (t >= ESL * HH) return;
  int e = t >> 2, h = t & 3;
  int s, d; edge_sd(ei, e, s, d);
  float v = lrelu(As[s * HH + h] + Ad[d * HH + h]);
  float m = dec_f(Mu[d * HH + h]);
  atomicAdd(&Den[d * HH + h], __expf(v - m));
}

// ---------------- edge pass 3: weighted message aggregation (bandwidth bound) ----------------
// one wave32 per (edge, head); lane owns 4 channels (float4 gather, 4 atomic f32 adds)
__global__ __launch_bounds__(256) void edge_agg_kernel(const int* __restrict__ ei,
                                                       const float* __restrict__ As,
                                                       const float* __restrict__ Ad,
                                                       const unsigned* __restrict__ Mu,
                                                       const float* __restrict__ Den,
                                                       const float* __restrict__ Hm,
                                                       float* __restrict__ Gout) {
  int gw = (int)(((size_t)blockIdx.x * blockDim.x + threadIdx.x) >> 5);
  int lane = threadIdx.x & 31;
  if (gw >= ESL * HH) return;
  int e = gw >> 2, h = gw & 3;
  int s, d; edge_sd(ei, e, s, d);
  float v = lrelu(As[s * HH + h] + Ad[d * HH + h]);
  float alpha = __expf(v - dec_f(Mu[d * HH + h])) / Den[d * HH + h];
  const float4 hv = ((const float4*)(Hm + (size_t)s * HC + h * CC))[lane];
  float* op = Gout + (size_t)d * HC + h * CC + lane * 4;
  atomicAdd(op + 0, alpha * hv.x);
  atomicAdd(op + 1, alpha * hv.y);
  atomicAdd(op + 2, alpha * hv.z);
  atomicAdd(op + 3, alpha * hv.w);
}

// ---------------- BN1 stats over biased GAT output ----------------
#define ROWS1 500
__global__ void bn1_partial_kernel(const float* __restrict__ G, const float* __restrict__ bias,
                                   float* __restrict__ sum, float* __restrict__ sq) {
  int half = blockIdx.x & 1;
  int chunk = blockIdx.x >> 1;
  int j = half * 256 + threadIdx.x;
  int r0 = chunk * ROWS1;
  int r1 = r0 + ROWS1; if (r1 > NN) r1 = NN;
  float b = bias[j];
  float s = 0.f, q = 0.f;
  for (int r = r0; r < r1; ++r) {
    float v = G[(size_t)r * HC + j] + b;
    s += v; q += v * v;
  }
  atomicAdd(&sum[j], s);
  atomicAdd(&sq[j], q);
}

__global__ void bn1_final_kernel(const float* __restrict__ sum, const float* __restrict__ sq,
                                 const float* __restrict__ gamma, const float* __restrict__ beta,
                                 const float* __restrict__ bias,
                                 float* __restrict__ scale, float* __restrict__ shift) {
  int j = threadIdx.x;  // HC threads
  float mean = sum[j] / (float)NN;
  float var  = sq[j] / (float)NN - mean * mean;
  float sc = gamma[j] * rsqrtf(var + BN_EPS);
  scale[j] = sc;
  shift[j] = sc * bias[j] + beta[j] - mean * sc;   // bias + BN affine folded
}

// ---------------- GEMM2: XW = relu(bn1(G)) [20000,512] @ gcn_W[512,128] (fp32 WMMA) ------------
// grid.x = 1250 (M tiles), block = 64 (2 waves); wave w covers cols [64w, 64w+64)
__global__ __launch_bounds__(64) void gemm2_kernel(const float* __restrict__ G,
                                                   const float* __restrict__ W2,
                                                   const float* __restrict__ scale,
                                                   const float* __restrict__ shift,
                                                   float* __restrict__ Y) {
  const int wave = threadIdx.x >> 5;
  const int lane = threadIdx.x & 31;
  const int lmod = lane & 15;
  const int lhi  = lane >> 4;
  const int m0 = blockIdx.x * 16;
  const int j0 = wave * 64;

  v8f c0 = {}, c1 = {}, c2 = {}, c3 = {};
  const float* arow = G + (size_t)(m0 + lmod) * HC;
  const int col = j0 + lmod;

  for (int kk = 0; kk < HC; kk += 4) {
    const int kA = kk + 2 * lhi;
    float a0 = scale[kA]     * arow[kA]     + shift[kA];
    float a1 = scale[kA + 1] * arow[kA + 1] + shift[kA + 1];
    v2f a; a[0] = a0 > 0.f ? a0 : 0.f; a[1] = a1 > 0.f ? a1 : 0.f;
    const float* b0p = W2 + (size_t)kA * CC;
    const float* b1p = W2 + (size_t)(kA + 1) * CC;
    v2f b;
    b[0] = b0p[col];      b[1] = b1p[col];
    c0 = __builtin_amdgcn_wmma_f32_16x16x4_f32(false, a, false, b, (short)0, c0, false, false);
    b[0] = b0p[col + 16]; b[1] = b1p[col + 16];
    c1 = __builtin_amdgcn_wmma_f32_16x16x4_f32(false, a, false, b, (short)0, c1, false, false);
    b[0] = b0p[col + 32]; b[1] = b1p[col + 32];
    c2 = __builtin_amdgcn_wmma_f32_16x16x4_f32(false, a, false, b, (short)0, c2, false, false);
    b[0] = b0p[col + 48]; b[1] = b1p[col + 48];
    c3 = __builtin_amdgcn_wmma_f32_16x16x4_f32(false, a, false, b, (short)0, c3, false, false);
  }
  const int row0 = m0 + 8 * lhi;
#pragma unroll
  for (int r = 0; r < 8; ++r) {
    float* o = Y + (size_t)(row0 + r) * CC + col;
    o[0]  = c0[r];
    o[16] = c1[r];
    o[32] = c2[r];
    o[48] = c3[r];
  }
}

// ---------------- GCN aggregation: one wave32 per edge ----------------
__global__ __launch_bounds__(256) void gcn_agg_kernel(const int* __restrict__ ei,
                                                      const float* __restrict__ deg,
                                                      const float* __restrict__ XW,
                                                      float* __restrict__ G2) {
  int gw = (int)(((size_t)blockIdx.x * blockDim.x + threadIdx.x) >> 5);
  int lane = threadIdx.x & 31;
  if (gw >= ESL) return;
  int s, d; edge_sd(ei, gw, s, d);
  float norm = rsqrtf(deg[s]) * rsqrtf(deg[d]);
  const float4 xv = ((const float4*)(XW + (size_t)s * CC))[lane];
  float* op = G2 + (size_t)d * CC + lane * 4;
  atomicAdd(op + 0, norm * xv.x);
  atomicAdd(op + 1, norm * xv.y);
  atomicAdd(op + 2, norm * xv.z);
  atomicAdd(op + 3, norm * xv.w);
}

// ---------------- BN2 stats ----------------
#define ROWS2 500
__global__ void bn2_partial_kernel(const float* __restrict__ G2, const float* __restrict__ bias,
                                   float* __restrict__ sum, float* __restrict__ sq) {
  int j = threadIdx.x;       // block = 128
  int r0 = blockIdx.x * ROWS2;
  int r1 = r0 + ROWS2; if (r1 > NN) r1 = NN;
  float b = bias[j];
  float s = 0.f, q = 0.f;
  for (int r = r0; r < r1; ++r) {
    float v = G2[(size_t)r * CC + j] + b;
    s += v; q += v * v;
  }
  atomicAdd(&sum[j], s);
  atomicAdd(&sq[j], q);
}

__global__ void bn2_final_kernel(const float* __restrict__ sum, const float* __restrict__ sq,
                                 const float* __restrict__ gamma, const float* __restrict__ beta,
                                 const float* __restrict__ bias,
                                 float* __restrict__ scale, float* __restrict__ shift) {
  int j = threadIdx.x;  // CC threads
  float mean = sum[j] / (float)NN;
  float var  = sq[j] / (float)NN - mean * mean;
  float sc = gamma[j] * rsqrtf(var + BN_EPS);
  scale[j] = sc;
  shift[j] = sc * bias[j] + beta[j] - mean * sc;
}

// ---------------- final: relu(bn2) + residual, / sqrt(2) ----------------
__global__ void final_kernel(const float* __restrict__ G2, const float* __restrict__ scale,
                             const float* __restrict__ shift, const float* __restrict__ X,
                             float* __restrict__ out) {
  int t = blockIdx.x * blockDim.x + threadIdx.x;
  if (t >= NN * CC) return;
  int c = t & (CC - 1);
  float v = scale[c] * G2[t] + shift[c];
  v = v > 0.f ? v : 0.f;
  out[t] = (v + X[t]) * 0.70710678118654752f;
}

// ---------------- host-side launcher ----------------
extern "C" void kernel_launch(void* const* d_in, const int* in_sizes, int n_in,
                              void* d_out, int out_size, void* d_ws, size_t ws_size,
                              hipStream_t stream) {
  const float* x        = (const float*)d_in[0];
  const int*   ei       = (const int*)d_in[1];
  const float* gat_W    = (const float*)d_in[2];
  const float* att_src  = (const float*)d_in[3];
  const float* att_dst  = (const float*)d_in[4];
  const float* gat_bias = (const float*)d_in[5];
  const float* bn1_g    = (const float*)d_in[6];
  const float* bn1_b    = (const float*)d_in[7];
  const float* gcn_W    = (const float*)d_in[8];
  const float* gcn_bias = (const float*)d_in[9];
  const float* bn2_g    = (const float*)d_in[10];
  const float* bn2_b    = (const float*)d_in[11];
  float* out = (float*)d_out;
  float* ws  = (float*)d_ws;

  float*    Hm   = ws + OFF_H;
  float*    Gat  = ws + OFF_GAT;
  float*    As   = ws + OFF_AS;
  float*    Ad   = ws + OFF_AD;
  unsigned* Mu   = (unsigned*)(ws + OFF_MU);
  float*    Den  = ws + OFF_DEN;
  float*    Deg  = ws + OFF_DEG;
  float*    Sum1 = ws + OFF_SUM1;
  float*    Sq1  = ws + OFF_SQ1;
  float*    Sc1  = ws + OFF_SC1;
  float*    Sh1  = ws + OFF_SH1;
  float*    XW   = ws + OFF_XW;
  float*    G2   = ws + OFF_GCN;
  float*    Sum2 = ws + OFF_SUM2;
  float*    Sq2  = ws + OFF_SQ2;
  float*    Sc2  = ws + OFF_SC2;
  float*    Sh2  = ws + OFF_SH2;

  // 1. re-initialize accumulators (deterministic across calls)
  zero_kernel<<<2048, 256, 0, stream>>>(Gat, OFF_XW - OFF_GAT);          // gat_out..shift1
  zero_kernel<<<1024, 256, 0, stream>>>(G2,  WS_FLOATS - OFF_GCN);       // gcn_out..shift2

  // 2. GEMM1 (WMMA f32): Hm = x @ gat_W
  gemm1_kernel<<<NN / 16, 256, 0, stream>>>(x, gat_W, Hm);

  // 3. attention logits per (node, head)
  alphas_kernel<<<(NN * HH + 255) / 256, 256, 0, stream>>>(Hm, att_src, att_dst, As, Ad);

  // 4-6. edge passes: segment max (+degree), softmax denom, weighted aggregation
  edge_max_kernel  <<<(ESL * HH + 255) / 256, 256, 0, stream>>>(ei, As, Ad, Mu, Deg);
  edge_denom_kernel<<<(ESL * HH + 255) / 256, 256, 0, stream>>>(ei, As, Ad, Mu, Den);
  edge_agg_kernel  <<<(ESL * HH * 32 + 255) / 256, 256, 0, stream>>>(ei, As, Ad, Mu, Den, Hm, Gat);

  // 7-8. BN1 statistics + fold (bias, gamma, beta, mean, var) into scale/shift
  bn1_partial_kernel<<<((NN + ROWS1 - 1) / ROWS1) * 2, 256, 0, stream>>>(Gat, gat_bias, Sum1, Sq1);
  bn1_final_kernel<<<1, HC, 0, stream>>>(Sum1, Sq1, bn1_g, bn1_b, gat_bias, Sc1, Sh1);

  // 9. GEMM2 (WMMA f32, BN1+ReLU fused into A-load): XW = relu(bn1(Gat)) @ gcn_W
  gemm2_kernel<<<NN / 16, 64, 0, stream>>>(Gat, gcn_W, Sc1, Sh1, XW);

  // 10. GCN normalized aggregation
  gcn_agg_kernel<<<(ESL * 32 + 255) / 256, 256, 0, stream>>>(ei, Deg, XW, G2);

  // 11-12. BN2 statistics + fold
  bn2_partial_kernel<<<(NN + ROWS2 - 1) / ROWS2, CC, 0, stream>>>(G2, gcn_bias, Sum2, Sq2);
  bn2_final_kernel<<<1, CC, 0, stream>>>(Sum2, Sq2, bn2_g, bn2_b, gcn_bias, Sc2, Sh2);

  // 13. relu(bn2) + residual, scaled by 1/sqrt(2)
  final_kernel<<<(NN * CC + 255) / 256, 256, 0, stream>>>(G2, Sc2, Sh2, x, out);
}